// GATEncoder_83021717831896
// MI455X (gfx1250) — compile-verified
//
#include <hip/hip_runtime.h>

// ---- problem constants (from reference) ----
#define N_NODES 50000
#define N_EDGES 800000
#define IN_CH   128
#define HID     64
#define HEADS   4
#define LATENT  64
#define HC1     (HEADS * HID)        // 256
#define E_TOT   (N_EDGES + N_NODES)  // 850000 (with self loops)
#define NEG_SLOPE 0.2f

typedef __attribute__((ext_vector_type(2))) float v2f;
typedef __attribute__((ext_vector_type(8))) float v8f;

// ---------------- utility ----------------
__global__ void fill_f32(float* __restrict__ p, float v, int n) {
    int t = blockIdx.x * blockDim.x + threadIdx.x;
    if (t < n) p[t] = v;
}

__device__ __forceinline__ void atomicMaxF(float* addr, float val) {
    int* ai = reinterpret_cast<int*>(addr);
    int cur = __float_as_int(*addr);
    while (__int_as_float(cur) < val) {
        int old = atomicCAS(ai, cur, __float_as_int(val));
        if (old == cur) break;
        cur = old;
    }
}

// ---------------- fp32 WMMA GEMM: C[M,N] = A[M,K] @ B[K,N] ----------------
// One wave per 16x16 output tile; K stepped by 4 via v_wmma_f32_16x16x4_f32.
// A frag: lanes 0-15 -> K={k,k+1}, lanes 16-31 -> K={k+2,k+3} (M = lane&15).
// B frag mirrored (N = lane&15). C/D: VGPR r -> row r (lanes<16) / r+8 (lanes>=16).
__global__ void wmma_gemm_f32(const float* __restrict__ A, const float* __restrict__ B,
                              float* __restrict__ C, int M, int N, int K) {
    const int lane   = threadIdx.x & 31;
    const int wave   = blockIdx.x * (blockDim.x >> 5) + (threadIdx.x >> 5);
    const int tilesN = N >> 4;
    const int tiles  = (M >> 4) * tilesN;
    if (wave >= tiles) return;                 // wave-uniform: EXEC stays all-ones
    const int tm  = wave / tilesN;
    const int tn  = wave - tm * tilesN;
    const int row = tm * 16 + (lane & 15);
    const int col = tn * 16 + (lane & 15);
    const int kh  = (lane >> 4) << 1;          // 0 or 2 depending on lane half

    v8f acc = {};
    const float* ap = A + (size_t)row * K + kh;
    for (int k = 0; k < K; k += 4) {
        v2f a = *reinterpret_cast<const v2f*>(ap + k);   // A[row][k+kh .. k+kh+1]
        v2f b;
        b.x = B[(size_t)(k + kh)     * N + col];
        b.y = B[(size_t)(k + kh + 1) * N + col];
        acc = __builtin_amdgcn_wmma_f32_16x16x4_f32(false, a, false, b,
                                                    (short)0, acc, false, false);
    }
    const int rbase = tm * 16 + ((lane >> 4) << 3);
#pragma unroll
    for (int r = 0; r < 8; ++r)
        C[(size_t)(rbase + r) * N + col] = acc[r];
}

// ---------------- attention coefficients: a_s[n,h] = <h[n,h,:], att_src[h,:]> ----------------
// one wave (32 lanes) per (node, head); C = 64 channels
__global__ void attn_coeff(const float* __restrict__ h, const float* __restrict__ att_src,
                           const float* __restrict__ att_dst, float* __restrict__ a_s,
                           float* __restrict__ a_d, int N, int H, int C) {
    const int lane = threadIdx.x & 31;
    const int wid  = blockIdx.x * (blockDim.x >> 5) + (threadIdx.x >> 5);
    if (wid >= N * H) return;
    const int n  = wid / H;
    const int hd = wid - n * H;
    const float* hp = h + (size_t)n * H * C + (size_t)hd * C;
    const float* as = att_src + (size_t)hd * C;
    const float* ad = att_dst + (size_t)hd * C;
    float ss = 0.f, sd = 0.f;
    for (int c = lane; c < C; c += 32) {
        float v = hp[c];
        ss += v * as[c];
        sd += v * ad[c];
    }
#pragma unroll
    for (int off = 16; off; off >>= 1) {
        ss += __shfl_down(ss, off);
        sd += __shfl_down(sd, off);
    }
    if (lane == 0) { a_s[wid] = ss; a_d[wid] = sd; }
}

__device__ __forceinline__ void decode_edge(const int* __restrict__ ei, int e,
                                            int E, int& s, int& d) {
    if (e < E) { s = ei[e]; d = ei[E + e]; }
    else       { s = d = e - E; }  // self loop
}

// pass 1: segment max of leaky_relu(a_s[src]+a_d[dst]) over dst
__global__ void edge_max(const int* __restrict__ ei, const float* __restrict__ a_s,
                         const float* __restrict__ a_d, float* __restrict__ m,
                         int E, int N, int H) {
    int t = blockIdx.x * blockDim.x + threadIdx.x;
    int total = (E + N) * H;
    if (t >= total) return;
    int e = t / H, hd = t - e * H, s, d;
    decode_edge(ei, e, E, s, d);
    float x = a_s[s * H + hd] + a_d[d * H + hd];
    float lr = x > 0.f ? x : NEG_SLOPE * x;
    atomicMaxF(m + d * H + hd, lr);
}

// pass 2: numerators (cached in alpha) + segment-sum denominator
__global__ void edge_exp(const int* __restrict__ ei, const float* __restrict__ a_s,
                         const float* __restrict__ a_d, const float* __restrict__ m,
                         float* __restrict__ den, float* __restrict__ alpha,
                         int E, int N, int H) {
    int t = blockIdx.x * blockDim.x + threadIdx.x;
    int total = (E + N) * H;
    if (t >= total) return;
    int e = t / H, hd = t - e * H, s, d;
    decode_edge(ei, e, E, s, d);
    float x = a_s[s * H + hd] + a_d[d * H + hd];
    float lr = x > 0.f ? x : NEG_SLOPE * x;
    float ex = expf(lr - m[d * H + hd]);
    alpha[t] = ex;
    atomicAdd(den + d * H + hd, ex);
}

// pass 3: out[dst] += (alpha/den[dst]) * h[src]; one thread per (edge, channel)
// HC1=256 -> one block per edge: fully coalesced gather + atomic scatter
__global__ void edge_agg(const int* __restrict__ ei, const float* __restrict__ h,
                         const float* __restrict__ alpha, const float* __restrict__ den,
                         float* __restrict__ out, int E, int N, int H, int C) {
    int t = blockIdx.x * blockDim.x + threadIdx.x;
    const int HC = H * C;
    int total = (E + N) * HC;  // <= 217.6M, fits in int
    if (t >= total) return;
    int e  = t / HC;
    int ch = t - e * HC;
    int hd = ch / C;
    int s, d;
    decode_edge(ei, e, E, s, d);
    float al = alpha[e * H + hd] / den[d * H + hd];
    atomicAdd(out + (size_t)d * HC + ch, al * h[(size_t)s * HC + ch]);
}

// layer-1 epilogue: out = elu(out + b) in place
__global__ void bias_elu(float* __restrict__ out, const float* __restrict__ b, int N, int HC) {
    int t = blockIdx.x * blockDim.x + threadIdx.x;
    if (t >= N * HC) return;
    float v = out[t] + b[t % HC];
    out[t] = v > 0.f ? v : (expf(v) - 1.f);
}

// final epilogue: out += b
__global__ void bias_add(float* __restrict__ out, const float* __restrict__ b, int N, int C) {
    int t = blockIdx.x * blockDim.x + threadIdx.x;
    if (t >= N * C) return;
    out[t] += b[t % C];
}

// ---------------- host orchestration ----------------
static inline int cdiv(int a, int b) { return (a + b - 1) / b; }

extern "C" void kernel_launch(void* const* d_in, const int* in_sizes, int n_in,
                              void* d_out, int out_size, void* d_ws, size_t ws_size,
                              hipStream_t stream) {
    (void)in_sizes; (void)n_in; (void)out_size; (void)ws_size;

    const float* x        = (const float*)d_in[0];
    const int*   ei       = (const int*)  d_in[1];   // [2, E] (row0 = src, row1 = dst)
    const float* W1       = (const float*)d_in[2];
    const float* att_s1   = (const float*)d_in[3];
    const float* att_d1   = (const float*)d_in[4];
    const float* b1       = (const float*)d_in[5];
    const float* W_mu     = (const float*)d_in[6];
    const float* att_s_mu = (const float*)d_in[7];
    const float* att_d_mu = (const float*)d_in[8];
    const float* b_mu     = (const float*)d_in[9];
    const float* W_ls     = (const float*)d_in[10];
    const float* att_s_ls = (const float*)d_in[11];
    const float* att_d_ls = (const float*)d_in[12];
    const float* b_ls     = (const float*)d_in[13];

    // workspace layout (floats)
    float* ws    = (float*)d_ws;
    float* h1    = ws;                       // 12.8M  [N, 256]  (reused for hmu/hls later)
    float* out1  = ws + 12800000;            // 12.8M  [N, 256]
    float* alpha = ws + 25600000;            //  3.4M  [E_TOT, H] (reused per layer)
    float* a_s   = ws + 29000000;            //  200K
    float* a_d   = ws + 29200000;            //  200K
    float* m     = ws + 29400000;            //  200K
    float* den   = ws + 29600000;            //  200K

    const int TB = 256;
    const float NEG_INF = -3.402823466e38f;

    // ================= Layer 1: GATConv(128 -> 4 heads x 64, concat) + ELU =================
    fill_f32<<<cdiv(N_NODES * HEADS, TB), TB, 0, stream>>>(m,   NEG_INF, N_NODES * HEADS);
    fill_f32<<<cdiv(N_NODES * HEADS, TB), TB, 0, stream>>>(den, 0.f,     N_NODES * HEADS);
    fill_f32<<<cdiv(N_NODES * HC1,   TB), TB, 0, stream>>>(out1, 0.f,    N_NODES * HC1);

    // h1 = x @ W1   (one wave per 16x16 tile, 8 waves/block)
    {
        int tiles = (N_NODES / 16) * (HC1 / 16);
        wmma_gemm_f32<<<cdiv(tiles, 8), TB, 0, stream>>>(x, W1, h1, N_NODES, HC1, IN_CH);
    }
    attn_coeff<<<cdiv(N_NODES * HEADS, 8), TB, 0, stream>>>(h1, att_s1, att_d1, a_s, a_d,
                                                            N_NODES, HEADS, HID);
    edge_max<<<cdiv(E_TOT * HEADS, TB), TB, 0, stream>>>(ei, a_s, a_d, m, N_EDGES, N_NODES, HEADS);
    edge_exp<<<cdiv(E_TOT * HEADS, TB), TB, 0, stream>>>(ei, a_s, a_d, m, den, alpha,
                                                         N_EDGES, N_NODES, HEADS);
    edge_agg<<<cdiv(E_TOT * HC1, TB), TB, 0, stream>>>(ei, h1, alpha, den, out1,
                                                       N_EDGES, N_NODES, HEADS, HID);
    bias_elu<<<cdiv(N_NODES * HC1, TB), TB, 0, stream>>>(out1, b1, N_NODES, HC1);
    // out1 is now h = elu(gat1(x)), [N, 256]

    // ================= Layers 2/3: mu and logstd heads (1 head, 64 ch, mean==identity) ====
    float* hmu = ws;            // reuse h1 region: [N, 64]
    float* hls = ws + 3200000;  //                  [N, 64]
    {
        int tiles = (N_NODES / 16) * (LATENT / 16);
        wmma_gemm_f32<<<cdiv(tiles, 8), TB, 0, stream>>>(out1, W_mu, hmu, N_NODES, LATENT, HC1);
        wmma_gemm_f32<<<cdiv(tiles, 8), TB, 0, stream>>>(out1, W_ls, hls, N_NODES, LATENT, HC1);
    }

    float* out_mu = (float*)d_out;            // [N, 64]
    float* out_ls = out_mu + N_NODES * LATENT;

    // ---- mu head ----
    fill_f32<<<cdiv(N_NODES, TB), TB, 0, stream>>>(m,   NEG_INF, N_NODES);
    fill_f32<<<cdiv(N_NODES, TB), TB, 0, stream>>>(den, 0.f,     N_NODES);
    fill_f32<<<cdiv(N_NODES * LATENT, TB), TB, 0, stream>>>(out_mu, 0.f, N_NODES * LATENT);
    attn_coeff<<<cdiv(N_NODES, 8), TB, 0, stream>>>(hmu, att_s_mu, att_d_mu, a_s, a_d,
                                                    N_NODES, 1, LATENT);
    edge_max<<<cdiv(E_TOT, TB), TB, 0, stream>>>(ei, a_s, a_d, m, N_EDGES, N_NODES, 1);
    edge_exp<<<cdiv(E_TOT, TB), TB, 0, stream>>>(ei, a_s, a_d, m, den, alpha,
                                                 N_EDGES, N_NODES, 1);
    edge_agg<<<cdiv(E_TOT * LATENT, TB), TB, 0, stream>>>(ei, hmu, alpha, den, out_mu,
                                                          N_EDGES, N_NODES, 1, LATENT);
    bias_add<<<cdiv(N_NODES * LATENT, TB), TB, 0, stream>>>(out_mu, b_mu, N_NODES, LATENT);

    // ---- logstd head ----
    fill_f32<<<cdiv(N_NODES, TB), TB, 0, stream>>>(m,   NEG_INF, N_NODES);
    fill_f32<<<cdiv(N_NODES, TB), TB, 0, stream>>>(den, 0.f,     N_NODES);
    fill_f32<<<cdiv(N_NODES * LATENT, TB), TB, 0, stream>>>(out_ls, 0.f, N_NODES * LATENT);
    attn_coeff<<<cdiv(N_NODES, 8), TB, 0, stream>>>(hls, att_s_ls, att_d_ls, a_s, a_d,
                                                    N_NODES, 1, LATENT);
    edge_max<<<cdiv(E_TOT, TB), TB, 0, stream>>>(ei, a_s, a_d, m, N_EDGES, N_NODES, 1);
    edge_exp<<<cdiv(E_TOT, TB), TB, 0, stream>>>(ei, a_s, a_d, m, den, alpha,
                                                 N_EDGES, N_NODES, 1);
    edge_agg<<<cdiv(E_TOT * LATENT, TB), TB, 0, stream>>>(ei, hls, alpha, den, out_ls,
                                                          N_EDGES, N_NODES, 1, LATENT);
    bias_add<<<cdiv(N_NODES * LATENT, TB), TB, 0, stream>>>(out_ls, b_ls, N_NODES, LATENT);
}